// LoLAState_15607911154146
// MI455X (gfx1250) — compile-verified
//
#include <hip/hip_runtime.h>
#include <hip/hip_bf16.h>
#include <math.h>

// ---------------------------------------------------------------------------
// LoLA heap-state update for MI455X (gfx1250, wave32).
// B=2, C=1024, G=1024, H=32, D=128, F=128.
// Memory-bound (~310MB traffic -> ~13us at 23.3TB/s); GEMMs use
// V_WMMA_F32_16X16X4_F32 (native f32 WMMA, K=4).
// ---------------------------------------------------------------------------

#define BB 2
#define CC 1024
#define GG 1024
#define HH 32
#define DD 128
#define FF 128
#define EPSV 1e-6f
#define NTOK 2048   // G + C
#define LPAD 144    // padded LDS row stride (floats): 2-row stride = 32 banks mod 64

typedef float  v2f __attribute__((ext_vector_type(2)));
typedef float  v8f __attribute__((ext_vector_type(8)));

__device__ __forceinline__ v8f wmma_f32_k4(v2f a, v2f b, v8f c) {
    // D = A(16x4,f32) x B(4x16,f32) + C(16x16,f32)
    return __builtin_amdgcn_wmma_f32_16x16x4_f32(
        /*neg_a=*/false, a, /*neg_b=*/false, b,
        /*c_mod=*/(short)0, c, /*reuse_a=*/false, /*reuse_b=*/false);
}

// ---------------------------------------------------------------------------
// Kernel 1: heap_score[b,g,h] = || V_top[b,g,h,:] - (FK_top@H_sum)/(FK_top.S_sum+eps) ||
// grid = B*H*(G/16) blocks of 256 threads (8 waves).  Wave w owns N-tile w.
// Zs holds the per-row reciprocal 1/(Z+eps) so the per-element op is a mul.
// ---------------------------------------------------------------------------
__global__ void __launch_bounds__(256)
k_heap_score(const float* __restrict__ FK_top, const float* __restrict__ H_sum,
             const float* __restrict__ S_sum,  const float* __restrict__ V_top,
             float* __restrict__ heap_ws)
{
    __shared__ float Zinv[16];
    __shared__ float partial[8][16][16];

    const int blk = blockIdx.x;
    const int gt  = blk & 63;            // G/16 = 64 tiles
    const int bh  = blk >> 6;            // b*H + h
    const int h   = bh & (HH - 1);
    const int b   = bh >> 5;
    const int g0  = gt << 4;

    const int tid  = threadIdx.x;
    const int lane = tid & 31;
    const int w    = tid >> 5;           // wave id 0..7
    const int lm   = lane & 15;
    const int hi   = lane >> 4;          // 0 or 1

    // Zinv[m] = 1 / (FK_top[b,g0+m,h,:] . S_sum[b,h,:] + eps)
    if (tid < 16) {
        const float* fk = FK_top + (((size_t)(b * GG + g0 + tid) * HH + h) * FF);
        const float* ss = S_sum  + ((size_t)bh * FF);
        float z = 0.f;
        #pragma unroll 8
        for (int f = 0; f < FF; ++f) z += fk[f] * ss[f];
        Zinv[tid] = 1.0f / (z + EPSV);
    }
    __syncthreads();

    // GEMM: pred_num tile (16 x 16) at columns [w*16, w*16+16)
    const float* Abase = FK_top + (((size_t)(b * GG + g0 + lm) * HH + h) * FF) + 2 * hi;
    const float* Bbase = H_sum  + (((size_t)bh * FF) + 2 * hi) * DD + (w << 4) + lm;

    v8f acc = 0;
    #pragma unroll 8
    for (int f0 = 0; f0 < FF; f0 += 4) {
        v2f a;  a.x = Abase[f0];               a.y = Abase[f0 + 1];
        v2f bf; bf.x = Bbase[(size_t)f0 * DD]; bf.y = Bbase[(size_t)(f0 + 1) * DD];
        acc = wmma_f32_k4(a, bf, acc);
    }

    // squared diff vs V_top
    const float* Vb = V_top + (((size_t)(b * GG + g0) * HH + h) * DD) + (w << 4) + lm;
    #pragma unroll
    for (int i = 0; i < 8; ++i) {
        const int m = i + 8 * hi;
        const float pred = acc[i] * Zinv[m];
        const float diff = Vb[(size_t)m * HH * DD] - pred;
        partial[w][m][lm] = diff * diff;
    }
    __syncthreads();

    if (tid < 16) {
        float s = 0.f;
        #pragma unroll
        for (int ww = 0; ww < 8; ++ww)
            #pragma unroll
            for (int n = 0; n < 16; ++n) s += partial[ww][tid][n];
        heap_ws[(size_t)bh * GG + g0 + tid] = sqrtf(s);
    }
}

// ---------------------------------------------------------------------------
// Kernel 2: per-(b,h) bitonic sort of 2048 (score, idx) pairs, descending by
// score, ties broken by ascending idx (== stable argsort(-score)).
// grid = B*H blocks of 1024 threads.
// ---------------------------------------------------------------------------
__global__ void __launch_bounds__(1024)
k_sort(const float* __restrict__ score_c, const float* __restrict__ heap_ws,
       float* __restrict__ out_heap, int* __restrict__ top_idx,
       int* __restrict__ bot_idx)
{
    __shared__ float ss[NTOK];
    __shared__ int   si[NTOK];

    const int bh = blockIdx.x;
    const int h  = bh & (HH - 1);
    const int b  = bh >> 5;
    const int t  = threadIdx.x;

    ss[t]        = heap_ws[(size_t)bh * GG + t];
    si[t]        = t;
    ss[t + GG]   = score_c[((size_t)(b * CC + t)) * HH + h];
    si[t + GG]   = t + GG;
    __syncthreads();

    for (int k = 2; k <= NTOK; k <<= 1) {
        for (int j = k >> 1; j > 0; j >>= 1) {
            const int i = ((t & ~(j - 1)) << 1) | (t & (j - 1));
            const int l = i | j;
            const bool desc = ((i & k) == 0);
            const float sa = ss[i], sb = ss[l];
            const int   ia = si[i], ib = si[l];
            // strict total order: "i comes first" (bigger score, tie -> smaller idx)
            const bool iFirst = (sa > sb) || (sa == sb && ia < ib);
            if (desc ? !iFirst : iFirst) {
                ss[i] = sb; ss[l] = sa;
                si[i] = ib; si[l] = ia;
            }
            __syncthreads();
        }
    }

    // outputs
    out_heap[((size_t)(b * GG + t)) * HH + h] = ss[t];
    top_idx[(size_t)bh * GG + t] = si[t];
    bot_idx[(size_t)bh * GG + t] = si[GG + t];
}

// ---------------------------------------------------------------------------
// Kernel 3: gather top rows of cat_K / cat_V / cat_FK.  One wave per row,
// float4 per lane (row = 128 floats).  grid = 3*B*G*H/8 blocks of 256.
// ---------------------------------------------------------------------------
__global__ void __launch_bounds__(256)
k_gather(const int* __restrict__ top_idx,
         const float* __restrict__ K_top, const float* __restrict__ K_win,
         const float* __restrict__ V_top, const float* __restrict__ V_win,
         const float* __restrict__ FK_top, const float* __restrict__ FK_win,
         float* __restrict__ outK, float* __restrict__ outV,
         float* __restrict__ outFK)
{
    const int lane = threadIdx.x & 31;
    const int W    = blockIdx.x * 8 + (threadIdx.x >> 5);
    const int ROWS = BB * GG * HH;           // 65536
    const int tensor = W / ROWS;             // 0:K 1:V 2:FK
    const int r      = W - tensor * ROWS;
    const int b  = r >> 15;                  // / (G*H)
    const int g  = (r >> 5) & (GG - 1);
    const int h  = r & (HH - 1);

    const int idx = top_idx[((size_t)(b * HH + h)) * GG + g];

    const float* src_top; const float* src_win; float* dst;
    if (tensor == 0)      { src_top = K_top;  src_win = K_win;  dst = outK;  }
    else if (tensor == 1) { src_top = V_top;  src_win = V_win;  dst = outV;  }
    else                  { src_top = FK_top; src_win = FK_win; dst = outFK; }

    const float* src = (idx < GG)
        ? src_top + (((size_t)(b * GG + idx)      * HH + h) * DD)
        : src_win + (((size_t)(b * CC + idx - GG) * HH + h) * DD);
    float* d = dst + (((size_t)(b * GG + g) * HH + h) * DD);

    reinterpret_cast<float4*>(d)[lane] =
        reinterpret_cast<const float4*>(src)[lane];
}

// ---------------------------------------------------------------------------
// Kernel 4: H_sum += bot_FK^T @ bot_V  (128x128, K=1024 gathered tokens),
//           S_sum += sum(bot_FK).  One block (256 thr, 8 waves) per (b,h).
// Wave w computes the 16-row f-strip m in [16w,16w+16), all 8 d-tiles.
// Tokens staged 32 at a time into padded LDS (conflict-free fragment reads).
// ---------------------------------------------------------------------------
__global__ void __launch_bounds__(256)
k_update(const int* __restrict__ bot_idx,
         const float* __restrict__ V_top, const float* __restrict__ V_win,
         const float* __restrict__ FK_top, const float* __restrict__ FK_win,
         const float* __restrict__ H_sum, const float* __restrict__ S_sum,
         float* __restrict__ outH, float* __restrict__ outS)
{
    __shared__ float FKc[32][LPAD];
    __shared__ float Vc [32][LPAD];

    const int bh = blockIdx.x;
    const int h  = bh & (HH - 1);
    const int b  = bh >> 5;

    const int tid  = threadIdx.x;
    const int lane = tid & 31;
    const int w    = tid >> 5;
    const int lm   = lane & 15;
    const int hi   = lane >> 4;
    const int m0   = w << 4;

    v8f acc[8];
    #pragma unroll
    for (int i = 0; i < 8; ++i) acc[i] = 0;
    float sAcc = 0.f;

    for (int chunk = 0; chunk < 32; ++chunk) {
        // stage 32 gathered token rows (FK and V) into LDS
        #pragma unroll
        for (int rr = 0; rr < 4; ++rr) {
            const int q   = tid + 256 * rr;     // float4 slot 0..1023
            const int row = q >> 5;             // 32 float4 per row
            const int c4  = q & 31;
            const int idx = bot_idx[(size_t)bh * GG + chunk * 32 + row];
            const float* sFK = (idx < GG)
                ? FK_top + (((size_t)(b * GG + idx)      * HH + h) * FF)
                : FK_win + (((size_t)(b * CC + idx - GG) * HH + h) * FF);
            const float* sV = (idx < GG)
                ? V_top + (((size_t)(b * GG + idx)      * HH + h) * DD)
                : V_win + (((size_t)(b * CC + idx - GG) * HH + h) * DD);
            reinterpret_cast<float4*>(&FKc[row][0])[c4] =
                reinterpret_cast<const float4*>(sFK)[c4];
            reinterpret_cast<float4*>(&Vc[row][0])[c4] =
                reinterpret_cast<const float4*>(sV)[c4];
        }
        __syncthreads();

        // S_sum partial (threads 0..127 each own one f)
        if (tid < FF) {
            #pragma unroll 8
            for (int rr = 0; rr < 32; ++rr) sAcc += FKc[rr][tid];
        }

        // A[m][k] = FK[token k][f=m] ; B[k][n] = V[token k][d=n]
        #pragma unroll
        for (int kk = 0; kk < 32; kk += 4) {
            const int k0 = kk + 2 * hi;
            v2f a; a.x = FKc[k0][m0 + lm]; a.y = FKc[k0 + 1][m0 + lm];
            #pragma unroll
            for (int jn = 0; jn < 8; ++jn) {
                v2f bf; bf.x = Vc[k0][(jn << 4) + lm];
                        bf.y = Vc[k0 + 1][(jn << 4) + lm];
                acc[jn] = wmma_f32_k4(a, bf, acc[jn]);
            }
        }
        __syncthreads();
    }

    // write H_sum + acc
    const size_t hbase = (size_t)bh * FF * DD;
    #pragma unroll
    for (int jn = 0; jn < 8; ++jn) {
        #pragma unroll
        for (int i = 0; i < 8; ++i) {
            const int m = m0 + i + 8 * hi;
            const int n = (jn << 4) + lm;
            outH[hbase + (size_t)m * DD + n] =
                H_sum[hbase + (size_t)m * DD + n] + acc[jn][i];
        }
    }
    if (tid < FF)
        outS[(size_t)bh * FF + tid] = S_sum[(size_t)bh * FF + tid] + sAcc;
}

// ---------------------------------------------------------------------------
// launcher
// ---------------------------------------------------------------------------
extern "C" void kernel_launch(void* const* d_in, const int* in_sizes, int n_in,
                              void* d_out, int out_size, void* d_ws, size_t ws_size,
                              hipStream_t stream) {
    (void)in_sizes; (void)n_in; (void)out_size; (void)ws_size;

    // inputs (setup_inputs order)
    const float* score_c = (const float*)d_in[3];
    const float* K_win   = (const float*)d_in[4];
    const float* V_win   = (const float*)d_in[5];
    const float* FK_win  = (const float*)d_in[6];
    const float* K_top   = (const float*)d_in[7];
    const float* V_top   = (const float*)d_in[8];
    const float* FK_top  = (const float*)d_in[9];
    const float* H_sum   = (const float*)d_in[10];
    const float* S_sum   = (const float*)d_in[11];

    // outputs, concatenated flat in return order
    float* out      = (float*)d_out;
    float* out_heap = out;                                    // B*G*H      = 65536
    float* out_K    = out_heap + (size_t)BB * GG * HH;        // B*G*H*D
    float* out_V    = out_K    + (size_t)BB * GG * HH * DD;
    float* out_FK   = out_V    + (size_t)BB * GG * HH * DD;
    float* out_H    = out_FK   + (size_t)BB * GG * HH * FF;   // B*H*F*D
    float* out_S    = out_H    + (size_t)BB * HH * FF * DD;   // B*H*F

    // workspace: heap scores (f32), top idx, bot idx
    float* heap_ws = (float*)d_ws;                            // B*H*G
    int*   top_idx = (int*)(heap_ws + (size_t)BB * HH * GG);
    int*   bot_idx = top_idx + (size_t)BB * HH * GG;

    // 1) heap scores via f32 WMMA GEMM
    k_heap_score<<<BB * HH * (GG / 16), 256, 0, stream>>>(
        FK_top, H_sum, S_sum, V_top, heap_ws);

    // 2) per-(b,h) bitonic sort -> sorted scores, top/bot indices
    k_sort<<<BB * HH, 1024, 0, stream>>>(
        score_c, heap_ws, out_heap, top_idx, bot_idx);

    // 3) gather K/V/FK top rows
    k_gather<<<(3 * BB * GG * HH) / 8, 256, 0, stream>>>(
        top_idx, K_top, K_win, V_top, V_win, FK_top, FK_win,
        out_K, out_V, out_FK);

    // 4) H_sum / S_sum update via f32 WMMA GEMM over gathered bottom tokens
    k_update<<<BB * HH, 256, 0, stream>>>(
        bot_idx, V_top, V_win, FK_top, FK_win, H_sum, S_sum, out_H, out_S);
}